// EmbeddingLoss_55972013801905
// MI455X (gfx1250) — compile-verified
//
#include <hip/hip_runtime.h>
#include <math.h>

// Problem constants (match reference)
#define B_SZ   2048
#define N_SZ   16384
#define D_SZ   64
#define P_SZ   5
#define K_TOP  (B_SZ * 2)   // HARD_NEG_RATIO = 2 -> k = 4096

typedef __attribute__((ext_vector_type(2))) float v2f;
typedef __attribute__((ext_vector_type(8))) float v8f;

// Monotonic float <-> sortable uint key mapping
__device__ __forceinline__ unsigned f2key(float f) {
    unsigned u = __float_as_uint(f);
    return (u & 0x80000000u) ? ~u : (u | 0x80000000u);
}
__device__ __forceinline__ float key2f(unsigned k) {
    unsigned u = (k & 0x80000000u) ? (k ^ 0x80000000u) : ~k;
    return __uint_as_float(u);
}

// ---------------------------------------------------------------------------
// K0: item squared norms v2[n] = ||item[n]||^2 ; also zero the scalar output
// ---------------------------------------------------------------------------
__global__ void __launch_bounds__(256) k0_item_norms(
    const float* __restrict__ item, float* __restrict__ v2, float* __restrict__ out)
{
    int n = blockIdx.x * 256 + threadIdx.x;
    if (n == 0) out[0] = 0.0f;              // fresh accumulator every launch
    if (n < N_SZ) {
        const float* p = item + (size_t)n * D_SZ;
        float s = 0.0f;
        #pragma unroll
        for (int d = 0; d < D_SZ; ++d) s += p[d] * p[d];
        v2[n] = s;
    }
}

// ---------------------------------------------------------------------------
// K1: per-user: u2, pos_loss (diagonal, from UNMASKED loss), sign(t), |t|
// ---------------------------------------------------------------------------
__global__ void __launch_bounds__(256) k1_user_prep(
    const float* __restrict__ user, const float* __restrict__ item,
    const float* __restrict__ target,
    float* __restrict__ u2, float* __restrict__ pos_loss,
    float* __restrict__ sign_t, float* __restrict__ abs_t)
{
    int b = blockIdx.x * 256 + threadIdx.x;
    if (b >= B_SZ) return;
    const float* u = user + (size_t)b * D_SZ;
    const float* v = item + (size_t)b * D_SZ;   // diagonal pairs with item b
    float s = 0.0f, d2 = 0.0f;
    #pragma unroll
    for (int d = 0; d < D_SZ; ++d) {
        float ud = u[d];
        float df = ud - v[d];
        s  += ud * ud;
        d2 += df * df;
    }
    float t  = target[b];
    float sg = (t > 0.0f) ? 1.0f : ((t < 0.0f) ? -1.0f : 0.0f);
    u2[b]       = s;
    sign_t[b]   = sg;
    abs_t[b]    = fabsf(t);
    pos_loss[b] = -0.5f * fmaxf(d2, 0.0f) * sg;   // SIGMA = 1
}

// ---------------------------------------------------------------------------
// K2: fused WMMA GEMM (f32 16x16x4, D=64 -> 16 WMMAs per tile) + squared
// distance + sign + accidental-hit mask. One wave per 16x16 output tile;
// block = 8 waves covering a 16(M) x 128(N) tile. Writes masked losses
// (neg -> value, accidental -> -inf) into workspace [B][N].
// ---------------------------------------------------------------------------
__global__ void __launch_bounds__(256) k2_gemm_mask(
    const float* __restrict__ user, const float* __restrict__ item,
    const int*  __restrict__ item_idx, const int* __restrict__ pos_idx,
    const float* __restrict__ u2, const float* __restrict__ v2,
    const float* __restrict__ sign_t, float* __restrict__ masked)
{
    const int lane = threadIdx.x & 31;
    const int wave = threadIdx.x >> 5;
    const int m0   = blockIdx.y * 16;
    const int n0   = blockIdx.x * 128 + wave * 16;
    const int half = lane >> 4;     // K sub-pair: lanes 16-31 hold K+2,K+3
    const int r    = lane & 15;     // row (A) / col (B) within tile

    // A: 16x4 f32 tile of U rows; B: 4x16 tile of V^T (i.e. V rows as cols)
    const float* ua = user + (size_t)(m0 + r) * D_SZ + half * 2;
    const float* vb = item + (size_t)(n0 + r) * D_SZ + half * 2;

    v8f acc = {0.f, 0.f, 0.f, 0.f, 0.f, 0.f, 0.f, 0.f};
    #pragma unroll
    for (int k0 = 0; k0 < D_SZ / 4; ++k0) {
        v2f a, b;
        a.x = ua[k0 * 4];  a.y = ua[k0 * 4 + 1];
        b.x = vb[k0 * 4];  b.y = vb[k0 * 4 + 1];
        // (neg_a, A, neg_b, B, c_mod, C, reuse_a, reuse_b)
        acc = __builtin_amdgcn_wmma_f32_16x16x4_f32(
            false, a, false, b, (short)0, acc, false, false);
    }

    const int n     = n0 + r;       // C/D layout: N = lane % 16
    const int iid_n = item_idx[n];
    #pragma unroll
    for (int j = 0; j < 8; ++j) {
        const int m   = m0 + j + half * 8;   // C/D layout: VGPR j -> M = j (+8)
        float dot  = acc[j];
        float d2   = u2[m] + v2[n] - 2.0f * dot;
        float loss = -0.5f * fmaxf(d2, 0.0f) * sign_t[m];
        bool hit = (item_idx[m] == iid_n);
        #pragma unroll
        for (int p = 0; p < P_SZ; ++p)
            hit = hit || (pos_idx[m * P_SZ + p] == iid_n);
        masked[(size_t)m * N_SZ + n] = hit ? -INFINITY : loss;
    }
}

// ---------------------------------------------------------------------------
// K3: per row (one block/row, 256 threads): exact radix-select of the
// K_TOP-th largest masked loss (4 x 8-bit LDS histogram passes over
// monotonic keys), then logsumexp over {pos_loss} U {top-k values} with
// exact tie handling, then atomicAdd(ce * |t|) into out[0].
// Row data stays L2-resident (128MB < 192MB L2).
// ---------------------------------------------------------------------------
__global__ void __launch_bounds__(256) k3_select_lse(
    const float* __restrict__ masked, const float* __restrict__ pos_loss,
    const float* __restrict__ abs_t, float* __restrict__ out)
{
    __shared__ unsigned hist[256];
    __shared__ float    red[256];
    __shared__ unsigned sh_digit;
    __shared__ unsigned sh_krem;

    const int row = blockIdx.x;
    const int t   = threadIdx.x;
    const float* rp = masked + (size_t)row * N_SZ;

    // ---- 1) row max (for a stable logsumexp shift) ----
    unsigned lm = 0u;
    for (int n = t; n < N_SZ; n += 256) {
        unsigned k = f2key(rp[n]);
        lm = (k > lm) ? k : lm;
    }
    hist[t] = lm;
    __syncthreads();
    for (int s = 128; s > 0; s >>= 1) {
        if (t < s) hist[t] = (hist[t + s] > hist[t]) ? hist[t + s] : hist[t];
        __syncthreads();
    }
    const float pl   = pos_loss[row];
    const float mval = fmaxf(key2f(hist[0]), pl);   // finite: pl is finite
    __syncthreads();

    // ---- 2) radix select: exact key of the K_TOP-th largest element ----
    unsigned prefix = 0u, pmask = 0u;
    if (t == 0) sh_krem = (unsigned)K_TOP;
    __syncthreads();
    for (int pass = 0; pass < 4; ++pass) {
        const int shift = 24 - pass * 8;
        hist[t] = 0u;
        __syncthreads();
        for (int n = t; n < N_SZ; n += 256) {
            unsigned key = f2key(rp[n]);
            if ((key & pmask) == prefix)
                atomicAdd(&hist[(key >> shift) & 255u], 1u);
        }
        __syncthreads();
        if (t == 0) {
            unsigned krem = sh_krem, cum = 0u, bsel = 0u;
            for (int bn = 255; bn >= 0; --bn) {
                if (cum + hist[bn] >= krem) { bsel = (unsigned)bn; break; }
                cum += hist[bn];
            }
            sh_krem  = krem - cum;      // rank within the chosen bin
            sh_digit = bsel;
        }
        __syncthreads();
        prefix |= sh_digit << shift;
        pmask  |= 0xFFu << shift;
        __syncthreads();
    }
    const unsigned T    = prefix;       // exact key of k-th largest
    const float    tval = key2f(T);

    // ---- 3) sum exp over strictly-greater values; count them for ties ----
    float    lsum = 0.0f;
    unsigned lcnt = 0u;
    for (int n = t; n < N_SZ; n += 256) {
        float v = rp[n];
        if (f2key(v) > T) { lsum += __expf(v - mval); ++lcnt; }
    }
    red[t]  = lsum;
    hist[t] = lcnt;
    __syncthreads();
    for (int s = 128; s > 0; s >>= 1) {
        if (t < s) { red[t] += red[t + s]; hist[t] += hist[t + s]; }
        __syncthreads();
    }
    if (t == 0) {
        float    sum = red[0];
        unsigned cgt = hist[0];                       // count strictly > T
        float ties   = (float)((unsigned)K_TOP > cgt ? (unsigned)K_TOP - cgt : 0u);
        sum += ties * __expf(tval - mval);            // tval=-inf -> exp=0 -> 0
        sum += __expf(pl - mval);                     // the positive logit
        float ce = __logf(sum) + mval - pl;           // logsumexp - pos_loss
        atomicAdd(out, ce * abs_t[row]);
    }
}

// ---------------------------------------------------------------------------
// Launcher. Inputs: [0]=user_embed(B*D f32), [1]=item_embed(N*D f32),
// [2]=target(B f32), [3]=item_idx(N i32), [4]=pos_idx(B*P i32).
// Output: 1 float. Workspace: B*N + N + 4*B floats (~134.4 MB).
// ---------------------------------------------------------------------------
extern "C" void kernel_launch(void* const* d_in, const int* in_sizes, int n_in,
                              void* d_out, int out_size, void* d_ws, size_t ws_size,
                              hipStream_t stream) {
    const float* user   = (const float*)d_in[0];
    const float* item   = (const float*)d_in[1];
    const float* target = (const float*)d_in[2];
    const int*   iidx   = (const int*)d_in[3];
    const int*   pidx   = (const int*)d_in[4];
    float* out = (float*)d_out;

    float* ws      = (float*)d_ws;
    float* maskedL = ws;                                  // B*N
    float* v2      = maskedL + (size_t)B_SZ * N_SZ;       // N
    float* u2      = v2 + N_SZ;                           // B
    float* posl    = u2 + B_SZ;                           // B
    float* sgt     = posl + B_SZ;                         // B
    float* abt     = sgt + B_SZ;                          // B

    k0_item_norms<<<N_SZ / 256, 256, 0, stream>>>(item, v2, out);
    k1_user_prep<<<B_SZ / 256, 256, 0, stream>>>(user, item, target, u2, posl, sgt, abt);

    dim3 g2(N_SZ / 128, B_SZ / 16);   // (128, 128) blocks of 8 waves
    k2_gemm_mask<<<g2, 256, 0, stream>>>(user, item, iidx, pidx, u2, v2, sgt, maskedL);

    k3_select_lse<<<B_SZ, 256, 0, stream>>>(maskedL, posl, abt, out);
}